// RPNTargetLayer_22849226015386
// MI455X (gfx1250) — compile-verified
//
#include <hip/hip_runtime.h>
#include <cstdint>
#include <cstddef>

// Problem constants (match reference)
#define N_IMG 8
#define A_CNT 131072
#define T_CNT 64
#define NUM_TRAIN_C 256
#define K_POS 128
#define NBINS 1024
#define LIST_CAP 4096

// ---- CDNA5 feature probes (device pass only) ----
#if defined(__HIP_DEVICE_COMPILE__)
#if __has_builtin(__builtin_amdgcn_global_load_async_to_lds_b32)
#define HAVE_ASYNC_LDS 1
#endif
#if __has_builtin(__builtin_amdgcn_s_wait_asynccnt)
#define HAVE_WAIT_ASYNC 1
#endif
#endif

// addrspace-qualified pointer typedefs for the async builtin
typedef __attribute__((address_space(1))) int as1_int;
typedef __attribute__((address_space(3))) int as3_int;

// ---- workspace layout (bytes from d_ws). Total ~12.9 MB ----
static constexpr size_t OFF_BESTKEY = 0;                                    // u64 [N*T]
static constexpr size_t OFF_NPOS    = OFF_BESTKEY + (size_t)N_IMG*T_CNT*8;  // u32 [N]
static constexpr size_t OFF_NNEG    = OFF_NPOS + N_IMG*4;
static constexpr size_t OFF_LCP     = OFF_NNEG + N_IMG*4;                   // list counts
static constexpr size_t OFF_LCN     = OFF_LCP + N_IMG*4;
static constexpr size_t OFF_HISTP   = OFF_LCN + N_IMG*4;                    // u32 [N*NBINS]
static constexpr size_t OFF_HISTN   = OFF_HISTP + (size_t)N_IMG*NBINS*4;
static constexpr size_t OFF_BPOS    = OFF_HISTN + (size_t)N_IMG*NBINS*4;    // i32 [N]
static constexpr size_t OFF_NEEDP   = OFF_BPOS + N_IMG*4;
static constexpr size_t OFF_BNEG    = OFF_NEEDP + N_IMG*4;
static constexpr size_t OFF_NEEDN   = OFF_BNEG + N_IMG*4;
static constexpr size_t OFF_ZERO_END= OFF_NEEDN + N_IMG*4;
static constexpr size_t OFF_LISTP   = (OFF_ZERO_END + 255) & ~(size_t)255;  // u32 [N*LIST_CAP]
static constexpr size_t OFF_LISTN   = OFF_LISTP + (size_t)N_IMG*LIST_CAP*4;
static constexpr size_t OFF_MAXIOU  = OFF_LISTN + (size_t)N_IMG*LIST_CAP*4; // f32 [N*A]
static constexpr size_t OFF_BESTGT  = OFF_MAXIOU + (size_t)N_IMG*A_CNT*4;   // i32 [N*A]
static constexpr size_t OFF_CLASSES = OFF_BESTGT + (size_t)N_IMG*A_CNT*4;   // i32 [N*A]
static constexpr size_t WS_NEEDED   = OFF_CLASSES + (size_t)N_IMG*A_CNT*4;

__device__ __forceinline__ int binOf(float r) {
    int b = (int)(r * (float)NBINS);
    return b < 0 ? 0 : (b > NBINS - 1 ? NBINS - 1 : b);
}

// ---------------- K1: IoU, per-anchor row max/argmax, per-GT column argmax ----------------
__global__ __launch_bounds__(256) void k1_iou(
    const float* __restrict__ anchors, const float* __restrict__ tb,
    float* __restrict__ maxiou, int* __restrict__ bestgt, int* __restrict__ classes,
    unsigned long long* __restrict__ bestKey)
{
    __shared__ float sGT[T_CNT * 4];   // 64 boxes (y1,x1,y2,x2)
    __shared__ float sGA[T_CNT];       // gt areas

    const int tid  = threadIdx.x;
    const int gidx = blockIdx.x * 256 + tid;          // < N*A (exact grid)
    const int n    = gidx / A_CNT;
    const int a    = gidx % A_CNT;

    // Stage this image's 64 GT boxes (256 floats) into LDS, one b32 per thread.
    const float* g = tb + (size_t)n * T_CNT * 4;
#ifdef HAVE_ASYNC_LDS
    {
        // gfx1250 async copy: GLOBAL_LOAD_ASYNC_TO_LDS_B32, tracked by ASYNCcnt.
        // Builtin expects (addrspace(1) int*, addrspace(3) int*, imm offset, imm cpol).
        __builtin_amdgcn_global_load_async_to_lds_b32(
            (as1_int*)(uintptr_t)(g + tid),
            (as3_int*)(uint32_t)(uintptr_t)&sGT[tid],
            0, 0);
#ifdef HAVE_WAIT_ASYNC
        __builtin_amdgcn_s_wait_asynccnt(0);
#else
        asm volatile("s_wait_asynccnt 0" ::: "memory");
#endif
    }
#else
    sGT[tid] = g[tid];
#endif
    __syncthreads();
    if (tid < T_CNT) {
        float gy1 = sGT[tid*4+0], gx1 = sGT[tid*4+1];
        float gy2 = sGT[tid*4+2], gx2 = sGT[tid*4+3];
        sGA[tid] = (gy2 - gy1) * (gx2 - gx1);
    }
    __syncthreads();

    // Prefetch the anchor stream ahead (global_prefetch_b8).
    __builtin_prefetch((const char*)anchors + (size_t)gidx * 16 + 16384, 0, 1);

    const float4 av = ((const float4*)anchors)[gidx];
    float ay1 = av.x, ax1 = av.y, ay2 = av.z, ax2 = av.w;
    const bool inb = (ay1 >= 0.f) && (ax1 >= 0.f) && (ay2 <= 1.f) && (ax2 <= 1.f);
    if (!inb) { ay1 = ax1 = ay2 = ax2 = 0.f; }
    const float aarea = (ay2 - ay1) * (ax2 - ax1);

    float maxv = -1.0f;
    int   argt = 0;
    const unsigned int inv_idx = ~(unsigned int)a;   // larger == smaller anchor idx

    for (int t = 0; t < T_CNT; ++t) {
        float gy1 = sGT[t*4+0], gx1 = sGT[t*4+1];
        float gy2 = sGT[t*4+2], gx2 = sGT[t*4+3];
        float ih = fmaxf(fminf(ay2, gy2) - fmaxf(ay1, gy1), 0.f);
        float iw = fmaxf(fminf(ax2, gx2) - fmaxf(ax1, gx1), 0.f);
        float inter = ih * iw;
        float uni   = aarea + sGA[t] - inter;
        float iou   = (uni > 0.f) ? inter / uni : 0.f;
        if (iou > maxv) { maxv = iou; argt = t; }   // first-max tie-break (argmax axis=1)

        // per-GT column argmax: pack (iou_bits, ~idx) and wave32 max-reduce
        unsigned long long key =
            ((unsigned long long)__float_as_uint(iou) << 32) | (unsigned long long)inv_idx;
        #pragma unroll
        for (int off = 16; off > 0; off >>= 1) {
            unsigned long long o = __shfl_xor(key, off, 32);
            key = (o > key) ? o : key;
        }
        if ((tid & 31) == 0)
            atomicMax(&bestKey[n * T_CNT + t], key);
    }

    maxiou[gidx]  = maxv;
    bestgt[gidx]  = argt;
    classes[gidx] = (maxv < 0.3f) ? 0 : ((maxv >= 0.7f) ? 1 : -1);
}

// ---------------- K2: scatter best-anchor-per-GT -> class 1 ----------------
__global__ void k2_scatter(const unsigned long long* __restrict__ bestKey,
                           int* __restrict__ classes)
{
    int i = blockIdx.x * blockDim.x + threadIdx.x;     // N*T
    if (i >= N_IMG * T_CNT) return;
    int n = i / T_CNT;
    unsigned int inv = (unsigned int)(bestKey[i] & 0xFFFFFFFFull);
    int a = (int)(~inv);
    classes[(size_t)n * A_CNT + a] = 1;
}

// ---------------- K3: counts + 1024-bin histograms of rand values ----------------
__global__ __launch_bounds__(256) void k3_hist(
    const int* __restrict__ classes,
    const float* __restrict__ rpos, const float* __restrict__ rneg,
    unsigned* __restrict__ nPos, unsigned* __restrict__ nNeg,
    unsigned* __restrict__ histP, unsigned* __restrict__ histN)
{
    const int i = blockIdx.x * 256 + threadIdx.x;
    const int n = i / A_CNT;
    const int c = classes[i];
    const bool isp = (c == 1), isn = (c == 0);

    // wave32-aggregated counter updates
    unsigned long long mp = __ballot(isp);
    unsigned long long mn = __ballot(isn);
    if ((threadIdx.x & 31) == 0) {
        if (mp) atomicAdd(&nPos[n], (unsigned)__popcll(mp));
        if (mn) atomicAdd(&nNeg[n], (unsigned)__popcll(mn));
    }
    if (isp) atomicAdd(&histP[n * NBINS + binOf(rpos[i])], 1u);
    if (isn) atomicAdd(&histN[n * NBINS + binOf(rneg[i])], 1u);
}

// ---------------- K4: per-image boundary-bin scan ----------------
__global__ void k4_select(const unsigned* __restrict__ nPos, const unsigned* __restrict__ nNeg,
                          const unsigned* __restrict__ histP, const unsigned* __restrict__ histN,
                          int* __restrict__ Bpos, int* __restrict__ needPos,
                          int* __restrict__ Bneg, int* __restrict__ needNeg)
{
    const int n = blockIdx.x;
    if (threadIdx.x != 0) return;

    int np   = (int)nPos[n];
    int kpos = np < K_POS ? np : K_POS;
    if (np <= K_POS) { Bpos[n] = NBINS; needPos[n] = 0; }
    else {
        int cum = 0;
        for (int b = 0; b < NBINS; ++b) {
            int h = (int)histP[n * NBINS + b];
            if (cum + h > K_POS) { Bpos[n] = b; needPos[n] = K_POS - cum; break; }
            cum += h;
        }
    }
    const int dn = NUM_TRAIN_C - kpos;      // desired negatives
    int nn = (int)nNeg[n];
    if (nn <= dn) { Bneg[n] = NBINS; needNeg[n] = 0; }
    else {
        int cum = 0;
        for (int b = 0; b < NBINS; ++b) {
            int h = (int)histN[n * NBINS + b];
            if (cum + h > dn) { Bneg[n] = b; needNeg[n] = dn - cum; break; }
            cum += h;
        }
    }
}

// ---------------- K5: gather boundary-bin members ----------------
__global__ __launch_bounds__(256) void k5_list(
    const int* __restrict__ classes,
    const float* __restrict__ rpos, const float* __restrict__ rneg,
    const int* __restrict__ Bpos, const int* __restrict__ Bneg,
    unsigned* __restrict__ lcp, unsigned* __restrict__ lcn,
    unsigned* __restrict__ listP, unsigned* __restrict__ listN)
{
    const int i = blockIdx.x * 256 + threadIdx.x;
    const int n = i / A_CNT;
    const int a = i % A_CNT;
    const int c = classes[i];
    if (c == 1) {
        if (binOf(rpos[i]) == Bpos[n]) {
            unsigned p = atomicAdd(&lcp[n], 1u);
            if (p < LIST_CAP) listP[(size_t)n * LIST_CAP + p] = (unsigned)a;
        }
    } else if (c == 0) {
        if (binOf(rneg[i]) == Bneg[n]) {
            unsigned p = atomicAdd(&lcn[n], 1u);
            if (p < LIST_CAP) listN[(size_t)n * LIST_CAP + p] = (unsigned)a;
        }
    }
}

// ---------------- K6: finalize classes (exact ranks in boundary bin) + deltas ----------------
__global__ __launch_bounds__(256) void k6_final(
    const float* __restrict__ anchors, const float* __restrict__ tb,
    const int* __restrict__ classes, const int* __restrict__ bestgt,
    const float* __restrict__ rpos, const float* __restrict__ rneg,
    const int* __restrict__ Bpos, const int* __restrict__ needPos,
    const int* __restrict__ Bneg, const int* __restrict__ needNeg,
    const unsigned* __restrict__ lcp, const unsigned* __restrict__ lcn,
    const unsigned* __restrict__ listP, const unsigned* __restrict__ listN,
    int* __restrict__ outC, float* __restrict__ outD)
{
    const int i = blockIdx.x * 256 + threadIdx.x;
    const int n = i / A_CNT;
    const int a = i % A_CNT;
    int c = classes[i];

    if (c == 1) {
        const float r = rpos[i];
        const int b = binOf(r), B = Bpos[n];
        if (b > B) c = -1;
        else if (b == B) {
            const int m = (int)min(lcp[n], (unsigned)LIST_CAP);
            const unsigned* L = &listP[(size_t)n * LIST_CAP];
            const float*    R = &rpos[(size_t)n * A_CNT];
            int rank = 0;
            for (int e = 0; e < m; ++e) {
                unsigned ea = L[e];
                float re = R[ea];
                rank += (re < r) || (re == r && (int)ea < a);
            }
            if (rank >= needPos[n]) c = -1;
        }
    } else if (c == 0) {
        const float r = rneg[i];
        const int b = binOf(r), B = Bneg[n];
        if (b > B) c = -1;
        else if (b == B) {
            const int m = (int)min(lcn[n], (unsigned)LIST_CAP);
            const unsigned* L = &listN[(size_t)n * LIST_CAP];
            const float*    R = &rneg[(size_t)n * A_CNT];
            int rank = 0;
            for (int e = 0; e < m; ++e) {
                unsigned ea = L[e];
                float re = R[ea];
                rank += (re < r) || (re == r && (int)ea < a);
            }
            if (rank >= needNeg[n]) c = -1;
        }
    }
    outC[i] = c;

    float4 dv = make_float4(0.f, 0.f, 0.f, 0.f);
    if (c > 0) {
        float4 av = ((const float4*)anchors)[i];
        float ay1 = av.x, ax1 = av.y, ay2 = av.z, ax2 = av.w;
        if (!((ay1 >= 0.f) && (ax1 >= 0.f) && (ay2 <= 1.f) && (ax2 <= 1.f)))
            ay1 = ax1 = ay2 = ax2 = 0.f;
        const int bg = bestgt[i];
        const float* gp = tb + ((size_t)n * T_CNT + bg) * 4;
        float gy1 = gp[0], gx1 = gp[1], gy2 = gp[2], gx2 = gp[3];

        float ah = ay2 - ay1, aw = ax2 - ax1;
        float acy = ay1 + 0.5f * ah, acx = ax1 + 0.5f * aw;
        float gh = gy2 - gy1, gw = gx2 - gx1;
        float gcy = gy1 + 0.5f * gh, gcx = gx1 + 0.5f * gw;
        float ahs = ah > 0.f ? ah : 1.f, aws = aw > 0.f ? aw : 1.f;
        float ghs = gh > 0.f ? gh : 1.f, gws = gw > 0.f ? gw : 1.f;
        dv = make_float4((gcy - acy) / ahs, (gcx - acx) / aws,
                         logf(ghs / ahs), logf(gws / aws));
    }
    ((float4*)outD)[i] = dv;
}

// ---------------- host launcher ----------------
extern "C" void kernel_launch(void* const* d_in, const int* in_sizes, int n_in,
                              void* d_out, int out_size, void* d_ws, size_t ws_size,
                              hipStream_t stream)
{
    (void)in_sizes; (void)n_in; (void)out_size; (void)ws_size; // WS_NEEDED ~12.9 MB

    const float* anchors = (const float*)d_in[0];
    const float* tb      = (const float*)d_in[1];
    const float* rpos    = (const float*)d_in[2];
    const float* rneg    = (const float*)d_in[3];

    char* ws = (char*)d_ws;
    unsigned long long* bestKey = (unsigned long long*)(ws + OFF_BESTKEY);
    unsigned* nPos   = (unsigned*)(ws + OFF_NPOS);
    unsigned* nNeg   = (unsigned*)(ws + OFF_NNEG);
    unsigned* lcp    = (unsigned*)(ws + OFF_LCP);
    unsigned* lcn    = (unsigned*)(ws + OFF_LCN);
    unsigned* histP  = (unsigned*)(ws + OFF_HISTP);
    unsigned* histN  = (unsigned*)(ws + OFF_HISTN);
    int* Bpos        = (int*)(ws + OFF_BPOS);
    int* needP       = (int*)(ws + OFF_NEEDP);
    int* Bneg        = (int*)(ws + OFF_BNEG);
    int* needN       = (int*)(ws + OFF_NEEDN);
    unsigned* listP  = (unsigned*)(ws + OFF_LISTP);
    unsigned* listN  = (unsigned*)(ws + OFF_LISTN);
    float* maxiou    = (float*)(ws + OFF_MAXIOU);
    int* bestgt      = (int*)(ws + OFF_BESTGT);
    int* classes     = (int*)(ws + OFF_CLASSES);

    int*   outC = (int*)d_out;                           // classes as int32 bits
    float* outD = (float*)d_out + (size_t)N_IMG * A_CNT; // deltas [N,A,4]

    const int TOT = N_IMG * A_CNT;
    const int NB  = TOT / 256;

    (void)hipMemsetAsync(ws, 0, OFF_ZERO_END, stream);

    k1_iou<<<NB, 256, 0, stream>>>(anchors, tb, maxiou, bestgt, classes, bestKey);
    k2_scatter<<<(N_IMG * T_CNT + 255) / 256, 256, 0, stream>>>(bestKey, classes);
    k3_hist<<<NB, 256, 0, stream>>>(classes, rpos, rneg, nPos, nNeg, histP, histN);
    k4_select<<<N_IMG, 1, 0, stream>>>(nPos, nNeg, histP, histN, Bpos, needP, Bneg, needN);
    k5_list<<<NB, 256, 0, stream>>>(classes, rpos, rneg, Bpos, Bneg, lcp, lcn, listP, listN);
    k6_final<<<NB, 256, 0, stream>>>(anchors, tb, classes, bestgt, rpos, rneg,
                                     Bpos, needP, Bneg, needN, lcp, lcn, listP, listN,
                                     outC, outD);
}